// TitansMemory_58351425683676
// MI455X (gfx1250) — compile-verified
//
#include <hip/hip_runtime.h>
#include <hip/hip_bf16.h>
#include <math.h>
#include <stdint.h>

// Problem constants (match reference setup_inputs)
#define B_Q    64
#define S_Q    128
#define D_DIM  256
#define CAP    500000
#define TOPK   5
#define CHUNK  1024
#define KP_ITER 32                        // keys staged per pass (2 tiles of 16)
#define ITERS  (CHUNK / KP_ITER)          // 32
#define NCHUNK ((CAP + CHUNK - 1) / CHUNK) // 489
#define NEG_BIG (-3.402823466e+38f)
#define KPITCH 260                         // LDS floats per staged key row (bank-conflict-free)
#define SIMP   33                          // sims row pitch (conflict-free column scans)

typedef float v2f __attribute__((ext_vector_type(2)));
typedef float v8f __attribute__((ext_vector_type(8)));

// ---------------------------------------------------------------------------
// Kernel 1: q = mean_S(query); q_norm = q / max(||q||, 1e-12)   -> ws
// ---------------------------------------------------------------------------
__global__ void titans_qnorm_kernel(const float* __restrict__ query,
                                    float* __restrict__ qn) {
  __shared__ float red[256];
  const int b = blockIdx.x;
  const int d = threadIdx.x;
  const float* qb = query + (size_t)b * S_Q * D_DIM + d;
  float s = 0.f;
#pragma unroll 4
  for (int t = 0; t < S_Q; ++t) s += qb[(size_t)t * D_DIM];
  const float m = s * (1.0f / (float)S_Q);
  red[d] = m * m;
  __syncthreads();
  for (int off = 128; off > 0; off >>= 1) {
    if (d < off) red[d] += red[d + off];
    __syncthreads();
  }
  const float scale = 1.0f / fmaxf(sqrtf(red[0]), 1e-12f);
  qn[(size_t)b * D_DIM + d] = m * scale;
}

// ---------------------------------------------------------------------------
// Kernel 2: initialize counts output = access_counts (d_out is poisoned)
// ---------------------------------------------------------------------------
__global__ void titans_copy_counts_kernel(const float* __restrict__ src,
                                          float* __restrict__ dst, int n) {
  const int i = blockIdx.x * blockDim.x + threadIdx.x;
  if (i < n) dst[i] = src[i];
}

// ---------------------------------------------------------------------------
// Kernel 3: cosine sims via V_WMMA_F32_16X16X4_F32, keys staged to LDS with
// gfx1250 async copies (global_load_async_to_lds_b128 + s_wait_asynccnt),
// double-buffered; per-chunk top-5 per query row.
// Grid: NCHUNK blocks x 256 threads (8 waves = 4 m-tiles x 2 n-tiles).
// ---------------------------------------------------------------------------
__global__ void __launch_bounds__(256, 1)
titans_sim_topk_kernel(const float* __restrict__ qn,
                       const float* __restrict__ keys,
                       float* __restrict__ cval,
                       int* __restrict__ cidx) {
  __shared__ float kbuf[2][KP_ITER * KPITCH];  // 2 x 33.3 KB staged keys
  __shared__ float sims[B_Q * SIMP];           // 8.25 KB

  const int tid  = threadIdx.x;
  const int wave = tid >> 5;
  const int lane = tid & 31;
  const int l16  = lane & 15;
  const int half = lane >> 4;
  const int mt   = wave & 3;   // query tile 0..3 (16 queries each)
  const int nt   = wave >> 2;  // key subtile 0..1 (16 keys each)
  const int chunk_start = blockIdx.x * CHUNK;

  // LDS byte offsets of the two staging buffers (generic-ptr low 32 bits).
  const uint32_t kb_off0 = (uint32_t)(uintptr_t)&kbuf[0][0];
  const uint32_t kb_off1 = (uint32_t)(uintptr_t)&kbuf[1][0];

  // Issue one stage: 32 keys x 256 f32, 16B per lane, coalesced rows.
  auto stage = [&](int sit, int bufi) {
    const int n0 = chunk_start + sit * KP_ITER;
    const uint32_t base = bufi ? kb_off1 : kb_off0;
#pragma unroll
    for (int p = 0; p < 8; ++p) {
      const int c   = tid + p * 256;   // chunk id 0..2047 (16B chunks)
      const int kl  = c >> 6;          // local key row (64 chunks per row)
      const int col = (c & 63) << 2;   // starting float column
      int grow = n0 + kl;
      grow = (grow < CAP) ? grow : (CAP - 1);     // branchless tail clamp
      const uint64_t g = (uint64_t)(uintptr_t)(keys + (size_t)grow * D_DIM + col);
      const uint32_t l = base + (uint32_t)((kl * KPITCH + col) * 4);
      asm volatile("global_load_async_to_lds_b128 %0, %1, off"
                   :: "v"(l), "v"(g) : "memory");
    }
  };

  // Load this wave's A fragments once: 64 ksteps x float2 = 128 VGPRs.
  // A 16x4 f32 layout: lanes 0-15 M=lane K={0,1}; lanes 16-31 K={2,3}.
  v2f a[64];
  {
    const float* qp = qn + (size_t)(mt * 16 + l16) * D_DIM + half * 2;
#pragma unroll
    for (int ks = 0; ks < 64; ++ks) a[ks] = *(const v2f*)(qp + ks * 4);
  }

  float tv[TOPK];
  int   ti[TOPK];
#pragma unroll
  for (int j = 0; j < TOPK; ++j) { tv[j] = NEG_BIG; ti[j] = 0; }

  stage(0, 0);  // prime buffer 0

  for (int it = 0; it < ITERS; ++it) {
    const int cur = it & 1;
    if (it + 1 < ITERS) {
      stage(it + 1, cur ^ 1);  // overlap next stage with this pass's compute
      asm volatile("s_wait_asynccnt %0" :: "i"(8) : "memory");  // retire older 8
    } else {
      asm volatile("s_wait_asynccnt %0" :: "i"(0) : "memory");
    }
    __syncthreads();  // all waves' staging of kbuf[cur] visible

    // B 4x16 layout mirrors A: lane row = key (nt*16+l16), cols k0+half*2.
    const float* bb = &kbuf[cur][(nt * 16 + l16) * KPITCH + half * 2];
    v8f acc = {0.f, 0.f, 0.f, 0.f, 0.f, 0.f, 0.f, 0.f};
    float ss = 0.f;  // this half's partial sum-of-squares of the key row
#pragma unroll
    for (int ksg = 0; ksg < 8; ++ksg) {
      v2f bf[8];  // 8 ds_load_b64 in flight before use
#pragma unroll
      for (int j = 0; j < 8; ++j) bf[j] = *(const v2f*)(bb + (ksg * 8 + j) * 4);
#pragma unroll
      for (int j = 0; j < 8; ++j) {
        ss = fmaf(bf[j].x, bf[j].x, ss);
        ss = fmaf(bf[j].y, bf[j].y, ss);
        acc = __builtin_amdgcn_wmma_f32_16x16x4_f32(
            /*neg_a=*/false, a[ksg * 8 + j], /*neg_b=*/false, bf[j],
            /*c_mod=*/(short)0, acc, /*reuse_a=*/false, /*reuse_b=*/false);
      }
    }

    // Full ||key||^2 = this half's partial + partner half (same key row).
    const float sstot = ss + __shfl_xor(ss, 16, 32);
    const float scale = 1.0f / fmaxf(sqrtf(sstot), 1e-12f);
    const int   row   = chunk_start + it * KP_ITER + nt * 16 + l16;
    const bool  valid = (row < CAP);

    // C layout: vgpr r -> M = r + half*8, N = lane%16.
#pragma unroll
    for (int r = 0; r < 8; ++r) {
      const float v = valid ? acc[r] * scale : NEG_BIG;
      sims[(mt * 16 + half * 8 + r) * SIMP + nt * 16 + l16] = v;
    }
    __syncthreads();

    if (tid < B_Q) {
      const int qrow = tid;
#pragma unroll 4
      for (int j = 0; j < KP_ITER; ++j) {
        const float v = sims[qrow * SIMP + j];
        if (v > tv[TOPK - 1]) {
          const int idx = chunk_start + it * KP_ITER + j;
          int p = TOPK - 1;
          while (p > 0 && tv[p - 1] < v) {
            tv[p] = tv[p - 1]; ti[p] = ti[p - 1]; --p;
          }
          tv[p] = v; ti[p] = idx;
        }
      }
    }
    __syncthreads();  // also guards kbuf[cur] reuse two passes later
  }

  if (tid < B_Q) {
    const size_t base = ((size_t)blockIdx.x * B_Q + tid) * TOPK;
#pragma unroll
    for (int j = 0; j < TOPK; ++j) { cval[base + j] = tv[j]; cidx[base + j] = ti[j]; }
  }
}

// ---------------------------------------------------------------------------
// Kernel 4: merge per-chunk top-5 -> global top-5 per query row; gather
// values mean; scatter-add counts. One block per query row, 256 threads.
// ---------------------------------------------------------------------------
__global__ void titans_merge_gather_kernel(const float* __restrict__ cval,
                                           const int* __restrict__ cidx,
                                           const float* __restrict__ values,
                                           float* __restrict__ out_ret,
                                           float* __restrict__ out_counts) {
  __shared__ float lv[256 * TOPK];
  __shared__ int   li[256 * TOPK];
  __shared__ int   fi_sh[TOPK];

  const int row = blockIdx.x;
  const int tid = threadIdx.x;

  float tv[TOPK];
  int   ti[TOPK];
#pragma unroll
  for (int j = 0; j < TOPK; ++j) { tv[j] = NEG_BIG; ti[j] = 0; }

  const int NC = NCHUNK * TOPK;
  for (int i = tid; i < NC; i += 256) {
    const int c = i / TOPK, j = i % TOPK;
    const size_t off = ((size_t)c * B_Q + row) * TOPK + j;
    const float v = cval[off];
    if (v > tv[TOPK - 1]) {
      const int idx = cidx[off];
      int p = TOPK - 1;
      while (p > 0 && tv[p - 1] < v) { tv[p] = tv[p - 1]; ti[p] = ti[p - 1]; --p; }
      tv[p] = v; ti[p] = idx;
    }
  }
#pragma unroll
  for (int j = 0; j < TOPK; ++j) { lv[tid * TOPK + j] = tv[j]; li[tid * TOPK + j] = ti[j]; }
  __syncthreads();

  if (tid == 0) {
    float fv[TOPK]; int fid[TOPK];
#pragma unroll
    for (int j = 0; j < TOPK; ++j) { fv[j] = NEG_BIG; fid[j] = 0; }
    for (int i = 0; i < 256 * TOPK; ++i) {
      const float v = lv[i];
      if (v > fv[TOPK - 1]) {
        const int idx = li[i];
        int p = TOPK - 1;
        while (p > 0 && fv[p - 1] < v) { fv[p] = fv[p - 1]; fid[p] = fid[p - 1]; --p; }
        fv[p] = v; fid[p] = idx;
      }
    }
#pragma unroll
    for (int j = 0; j < TOPK; ++j) fi_sh[j] = fid[j];
  }
  __syncthreads();

  float s = 0.f;
#pragma unroll
  for (int j = 0; j < TOPK; ++j) s += values[(size_t)fi_sh[j] * D_DIM + tid];
  out_ret[(size_t)row * D_DIM + tid] = s * (1.0f / (float)TOPK);

  if (tid < TOPK) atomicAdd(&out_counts[fi_sh[tid]], 1.0f);
}

// ---------------------------------------------------------------------------
extern "C" void kernel_launch(void* const* d_in, const int* in_sizes, int n_in,
                              void* d_out, int out_size, void* d_ws, size_t ws_size,
                              hipStream_t stream) {
  (void)in_sizes; (void)n_in; (void)out_size; (void)ws_size;

  const float* query  = (const float*)d_in[0];  // [64,128,256]
  const float* keys   = (const float*)d_in[1];  // [500000,256]
  const float* values = (const float*)d_in[2];  // [500000,256]
  const float* counts = (const float*)d_in[3];  // [500000]

  float* out        = (float*)d_out;
  float* out_ret    = out;                       // [64,256]
  float* out_counts = out + (size_t)B_Q * D_DIM; // [500000]

  // Workspace layout (~1.26 MB):
  float* qn   = (float*)d_ws;                               // 64*256
  float* cval = qn + (size_t)B_Q * D_DIM;                   // NCHUNK*64*5
  int*   cidx = (int*)(cval + (size_t)NCHUNK * B_Q * TOPK); // NCHUNK*64*5

  titans_qnorm_kernel<<<B_Q, 256, 0, stream>>>(query, qn);
  titans_copy_counts_kernel<<<(CAP + 255) / 256, 256, 0, stream>>>(counts, out_counts, CAP);
  titans_sim_topk_kernel<<<NCHUNK, 256, 0, stream>>>(qn, keys, cval, cidx);
  titans_merge_gather_kernel<<<B_Q, 256, 0, stream>>>(cval, cidx, values, out_ret, out_counts);
}